// CIC_32899449487858
// MI455X (gfx1250) — compile-verified
//
#include <hip/hip_runtime.h>

typedef float v2f __attribute__((ext_vector_type(2)));
typedef float v8f __attribute__((ext_vector_type(8)));

#define BATCH 16
#define NPTS  4096
#define NCEN  1024
#define KNB   32
#define DIN   128
#define DOUT  256
#define DHID  64
#define SLOPE 0.01f

__device__ __forceinline__ float lrelu(float x) { return (x >= 0.f) ? x : SLOPE * x; }

// ---------------------------------------------------------------------------
// 1) Farthest point sampling: one block (1024 threads) per batch.
//    All 4096 point coords staged in LDS (48 KB of the 320 KB WGP LDS) so the
//    winning center's coords are readable by every thread without an extra
//    owner-publish barrier: 2 barriers per serial iteration instead of 3.
//    Writes p_ce (centers' coords) directly into the d_out tail.
// ---------------------------------------------------------------------------
__global__ void fps_kernel(const float* __restrict__ p, float* __restrict__ p_ce) {
  const int b    = blockIdx.x;
  const int tid  = threadIdx.x;          // 0..1023
  const int lane = tid & 31;
  const int wave = tid >> 5;             // 0..31

  const float* pb  = p    + (size_t)b * NPTS * 3;
  float*       pcb = p_ce + (size_t)b * NCEN * 3;

  __shared__ float sX[NPTS];             // 16 KB
  __shared__ float sY[NPTS];             // 16 KB
  __shared__ float sZ[NPTS];             // 16 KB
  __shared__ float sRedV[32];
  __shared__ int   sRedI[32];
  __shared__ int   sWin;

  float px[4], py[4], pz[4], dist[4];
#pragma unroll
  for (int q = 0; q < 4; ++q) {
    int i = tid + q * 1024;
    px[q] = pb[i * 3 + 0];
    py[q] = pb[i * 3 + 1];
    pz[q] = pb[i * 3 + 2];
    sX[i] = px[q]; sY[i] = py[q]; sZ[i] = pz[q];
    dist[q] = 1e10f;
  }
  if (tid == 0) {                        // center 0 = point 0
    pcb[0] = px[0]; pcb[1] = py[0]; pcb[2] = pz[0];
  }
  __syncthreads();
  float cx = sX[0], cy = sY[0], cz = sZ[0];

  for (int t = 1; t < NCEN; ++t) {
    float bestV = -1.0f; int bestI = 0x7fffffff;
#pragma unroll
    for (int q = 0; q < 4; ++q) {
      int i = tid + q * 1024;
      float dx = px[q] - cx, dy = py[q] - cy, dz = pz[q] - cz;
      float d = dx * dx + dy * dy + dz * dz;
      float nd = fminf(dist[q], d);
      dist[q] = nd;
      if (nd > bestV) { bestV = nd; bestI = i; }    // q ascending -> first max kept
    }
#pragma unroll
    for (int off = 16; off > 0; off >>= 1) {
      float oV = __shfl_xor(bestV, off, 32);
      int   oI = __shfl_xor(bestI, off, 32);
      if (oV > bestV || (oV == bestV && oI < bestI)) { bestV = oV; bestI = oI; }
    }
    if (lane == 0) { sRedV[wave] = bestV; sRedI[wave] = bestI; }
    __syncthreads();
    if (wave == 0) {
      float v = sRedV[lane]; int ii = sRedI[lane];
#pragma unroll
      for (int off = 16; off > 0; off >>= 1) {
        float oV = __shfl_xor(v, off, 32);
        int   oI = __shfl_xor(ii, off, 32);
        if (oV > v || (oV == v && oI < ii)) { v = oV; ii = oI; }
      }
      if (lane == 0) sWin = ii;
    }
    __syncthreads();
    const int w = sWin;
    cx = sX[w]; cy = sY[w]; cz = sZ[w];   // LDS broadcast read, no conflict
    if (tid == 0) {
      pcb[t * 3 + 0] = cx; pcb[t * 3 + 1] = cy; pcb[t * 3 + 2] = cz;
    }
  }
}

// ---------------------------------------------------------------------------
// 2) Ball query: one wave per center. First 32 in-radius indices in ascending
//    order (ballot + prefix popcount), tail filled with the first index.
// ---------------------------------------------------------------------------
__global__ void ball_kernel(const float* __restrict__ p,
                            const float* __restrict__ p_ce,
                            int* __restrict__ u_ne) {
  const int gwave = (blockIdx.x * blockDim.x + threadIdx.x) >> 5;
  const int lane  = threadIdx.x & 31;
  const int b = gwave / NCEN;
  const int j = gwave - b * NCEN;

  const float* pb = p + (size_t)b * NPTS * 3;
  const float* c  = p_ce + ((size_t)b * NCEN + j) * 3;
  const float cx = c[0], cy = c[1], cz = c[2];
  const float cn = cx * cx + cy * cy + cz * cz;
  const float R2 = (float)(0.2 * 0.2);
  int* out = u_ne + ((size_t)b * NCEN + j) * KNB;

  int cnt = 0, firstIdx = 0;
  for (int chunk = 0; chunk < NPTS / 32 && cnt < KNB; ++chunk) {
    int i = chunk * 32 + lane;
    float x = pb[i * 3], y = pb[i * 3 + 1], z = pb[i * 3 + 2];
    float d = cn + (x * x + y * y + z * z) - 2.0f * (cx * x + cy * y + cz * z);
    bool in = (d <= R2);                                    // ref: drop when d > R^2
    unsigned mask = (unsigned)__ballot(in);
    if (cnt == 0 && mask) firstIdx = chunk * 32 + __builtin_ctz(mask);
    if (in) {
      int slot = cnt + __popc(mask & ((1u << lane) - 1u));
      if (slot < KNB) out[slot] = i;
    }
    cnt += __popc(mask);
  }
  if (cnt < KNB && lane >= cnt) out[lane] = firstIdx;
}

// ---------------------------------------------------------------------------
// 3) Gather + max pool over 32 neighbors: one block (128 thr) per center.
// ---------------------------------------------------------------------------
__global__ void pool_kernel(const float* __restrict__ f,
                            const int* __restrict__ u_ne,
                            float* __restrict__ f_ce) {
  const int center = blockIdx.x;        // b*NCEN + j
  const int b = center >> 10;
  const int c = threadIdx.x;            // channel 0..127
  __shared__ int nbr[KNB];
  if (c < KNB) nbr[c] = u_ne[(size_t)center * KNB + c];
  __syncthreads();
  const float* fb = f + (size_t)b * NPTS * DIN;
  float m = -3.402823466e38f;
#pragma unroll 8
  for (int k = 0; k < KNB; ++k) m = fmaxf(m, fb[(size_t)nbr[k] * DIN + c]);
  f_ce[(size_t)center * DIN + c] = m;
}

// ---------------------------------------------------------------------------
// 4) Generic fp32 WMMA GEMM: D = act(A(MxK) @ W(KxN) + bias [+ Cres])
//    One wave per 16x16 output tile, K stepped by 4 (V_WMMA_F32_16X16X4_F32).
// ---------------------------------------------------------------------------
__global__ void gemm_wmma_kernel(const float* __restrict__ A,
                                 const float* __restrict__ W,
                                 const float* __restrict__ bias,
                                 const float* __restrict__ Cres,
                                 float* __restrict__ D,
                                 int N, int K, int act) {
  const int gwave  = (blockIdx.x * blockDim.x + threadIdx.x) >> 5;
  const int lane   = threadIdx.x & 31;
  const int ntiles = N >> 4;
  const int tm = gwave / ntiles;
  const int tn = gwave - tm * ntiles;

  const int r  = lane & 15;     // A row within tile / B-D column within tile
  const int kp = lane >> 4;     // K sub-pair: lanes 16-31 hold K=2,3

  const float* Arow = A + (size_t)(tm * 16 + r) * K;
  const float* Wcol = W + tn * 16 + r;

  v8f acc = {};
  for (int kb = 0; kb < K; kb += 4) {
    const int ka = kb + 2 * kp;                         // even -> 8B aligned
    v2f a = *(const v2f*)(Arow + ka);                   // A[r][ka], A[r][ka+1]
    v2f bb;
    bb.x = Wcol[(size_t)ka * N];
    bb.y = Wcol[(size_t)(ka + 1) * N];
    acc = __builtin_amdgcn_wmma_f32_16x16x4_f32(false, a, false, bb,
                                                (short)0, acc, false, false);
  }
  const int col = tn * 16 + r;
  const float bv = bias[col];
  const int rowBase = tm * 16 + (kp << 3);              // lanes 16-31 -> rows +8
#pragma unroll
  for (int v = 0; v < 8; ++v) {
    int row = rowBase + v;
    float x = acc[v] + bv;
    if (Cres) x += Cres[(size_t)row * N + col];
    if (act)  x = lrelu(x);
    D[(size_t)row * N + col] = x;
  }
}

// ---------------------------------------------------------------------------
// 5) kNN among centers: one wave per center; 1024 dists cached in 32 regs/lane;
//    32 argmin rounds with lexicographic (d, idx) tie-break (top_k semantics).
// ---------------------------------------------------------------------------
__global__ void knn_kernel(const float* __restrict__ p_ce, int* __restrict__ u_k) {
  const int gwave = (blockIdx.x * blockDim.x + threadIdx.x) >> 5;
  const int lane  = threadIdx.x & 31;
  const int b = gwave >> 10;
  const int j = gwave & 1023;
  const float* pcb = p_ce + (size_t)b * NCEN * 3;
  const float cx = pcb[j * 3], cy = pcb[j * 3 + 1], cz = pcb[j * 3 + 2];
  const float cn = cx * cx + cy * cy + cz * cz;

  float dloc[32];
#pragma unroll
  for (int c = 0; c < 32; ++c) {
    int i = c * 32 + lane;
    float x = pcb[i * 3], y = pcb[i * 3 + 1], z = pcb[i * 3 + 2];
    dloc[c] = cn + (x * x + y * y + z * z) - 2.0f * (cx * x + cy * y + cz * z);
  }
  int myOut = 0;
  for (int rnd = 0; rnd < KNB; ++rnd) {
    float bv = 3.2e38f; int bc = 0;
#pragma unroll
    for (int c = 0; c < 32; ++c)
      if (dloc[c] < bv) { bv = dloc[c]; bc = c; }       // first min kept (lowest idx)
    int bi = bc * 32 + lane;
#pragma unroll
    for (int off = 16; off > 0; off >>= 1) {
      float oV = __shfl_xor(bv, off, 32);
      int   oI = __shfl_xor(bi, off, 32);
      if (oV < bv || (oV == bv && oI < bi)) { bv = oV; bi = oI; }
    }
    if ((bi & 31) == lane) {
      int cw = bi >> 5;
#pragma unroll
      for (int c = 0; c < 32; ++c) if (c == cw) dloc[c] = 3.4e38f;
    }
    if (rnd == lane) myOut = bi;
  }
  u_k[(size_t)gwave * KNB + lane] = myOut;
}

// ---------------------------------------------------------------------------
// 6) Fused per-center MLP: G2 = lrelu(fh_ne - fh_c + p_cat@W_p2f + b) in LDS,
//    then WMMA f32 16x16x4 for G2(32x64) @ W_mlp(64x64), fused +bias, lrelu,
//    max over the 32 neighbors -> gmax (16384 x 64).
// ---------------------------------------------------------------------------
__global__ void center_kernel(const float* __restrict__ p_ce,
                              const float* __restrict__ fh,
                              const int*   __restrict__ u_k,
                              const float* __restrict__ W_p2f,
                              const float* __restrict__ b_p2f,
                              const float* __restrict__ W_mlp,
                              const float* __restrict__ b_mlp,
                              float* __restrict__ gmax) {
  const int center = blockIdx.x;        // b*NCEN + j
  const int b    = center >> 10;
  const int tid  = threadIdx.x;         // 0..127
  const int lane = tid & 31;
  const int wave = tid >> 5;

  __shared__ float sP[KNB][12];         // p_cat padded 9 -> 12 (cols 9..11 = 0)
  __shared__ float sG2[KNB][DHID];
  __shared__ float sFhC[DHID];
  __shared__ int   sNbr[KNB];
  __shared__ float sCen[3];

  if (tid < KNB)  sNbr[tid] = u_k[(size_t)center * KNB + tid];
  if (tid < DHID) sFhC[tid] = fh[(size_t)center * DHID + tid];
  if (tid < 3)    sCen[tid] = p_ce[(size_t)center * 3 + tid];
  __syncthreads();

  const float* pcb = p_ce + (size_t)b * NCEN * 3;
  if (tid < KNB) {
    int nb = sNbr[tid];
    float nx = pcb[nb * 3], ny = pcb[nb * 3 + 1], nz = pcb[nb * 3 + 2];
    float cx = sCen[0], cy = sCen[1], cz = sCen[2];
    sP[tid][0] = cx; sP[tid][1] = cy; sP[tid][2] = cz;
    sP[tid][3] = nx; sP[tid][4] = ny; sP[tid][5] = nz;
    sP[tid][6] = nx - cx; sP[tid][7] = ny - cy; sP[tid][8] = nz - cz;
    sP[tid][9] = 0.f; sP[tid][10] = 0.f; sP[tid][11] = 0.f;
  }
  __syncthreads();

  // G2: 32x64 elements, 128 threads -> 16 per thread
  {
    const int h = tid & 63;
    const int khalf = tid >> 6;
    const float* fhb = fh + (size_t)b * NCEN * DHID;
    const float bp = b_p2f[h];
    const float fc = sFhC[h];
#pragma unroll
    for (int kk = 0; kk < 16; ++kk) {
      int k = khalf * 16 + kk;
      int nb = sNbr[k];
      float acc = bp - fc + fhb[(size_t)nb * DHID + h];
#pragma unroll
      for (int q = 0; q < 9; ++q) acc += sP[k][q] * W_p2f[q * DHID + h];
      sG2[k][h] = lrelu(acc);
    }
  }
  __syncthreads();

  // WMMA: wave w owns cols n0..n0+15, both row tiles (k = 0..15 and 16..31)
  const int n0  = wave * 16;
  const int col = lane & 15;
  const int kp  = lane >> 4;
  const int r0  = lane & 15;
  v8f acc0 = {}, acc1 = {};
#pragma unroll
  for (int ks = 0; ks < DHID / 4; ++ks) {
    const int kb = ks * 4 + 2 * kp;                     // even -> aligned
    v2f a0 = *(const v2f*)(&sG2[r0][kb]);
    v2f a1 = *(const v2f*)(&sG2[16 + r0][kb]);
    v2f bb;
    bb.x = W_mlp[(size_t)kb * DHID + n0 + col];
    bb.y = W_mlp[(size_t)(kb + 1) * DHID + n0 + col];
    acc0 = __builtin_amdgcn_wmma_f32_16x16x4_f32(false, a0, false, bb,
                                                 (short)0, acc0, false, false);
    acc1 = __builtin_amdgcn_wmma_f32_16x16x4_f32(false, a1, false, bb,
                                                 (short)0, acc1, false, false);
  }
  // epilogue: +b_mlp, lrelu, max over all 32 neighbor rows
  const float bm = b_mlp[n0 + col];
  float m = -3.402823466e38f;
#pragma unroll
  for (int v = 0; v < 8; ++v) {
    float g0 = lrelu(acc0[v] + bm);
    float g1 = lrelu(acc1[v] + bm);
    m = fmaxf(m, fmaxf(g0, g1));
  }
  m = fmaxf(m, __shfl_xor(m, 16, 32));                  // rows v and v+8 halves
  if (lane < 16) gmax[(size_t)center * DHID + n0 + lane] = m;
}

// ---------------------------------------------------------------------------
extern "C" void kernel_launch(void* const* d_in, const int* in_sizes, int n_in,
                              void* d_out, int out_size, void* d_ws, size_t ws_size,
                              hipStream_t stream) {
  const float* f     = (const float*)d_in[0];
  const float* p     = (const float*)d_in[1];
  const float* W_sc  = (const float*)d_in[2];
  const float* b_sc  = (const float*)d_in[3];
  const float* W_pre = (const float*)d_in[4];
  const float* b_pre = (const float*)d_in[5];
  const float* W_p2f = (const float*)d_in[6];
  const float* b_p2f = (const float*)d_in[7];
  const float* W_mlp = (const float*)d_in[8];
  const float* b_mlp = (const float*)d_in[9];
  const float* W_pst = (const float*)d_in[10];
  const float* b_pst = (const float*)d_in[11];

  float* out  = (float*)d_out;                              // (B*NCEN, DOUT)
  float* p_ce = out + (size_t)BATCH * NCEN * DOUT;          // (B*NCEN, 3) output #2

  const size_t NC = (size_t)BATCH * NCEN;                   // 16384
  char* ws = (char*)d_ws;
  int*   u_ne = (int*)ws;    ws += NC * KNB * sizeof(int);    // 2 MB
  float* f_ce = (float*)ws;  ws += NC * DIN * sizeof(float);  // 8 MB
  float* res  = (float*)ws;  ws += NC * DOUT * sizeof(float); // 16 MB
  float* fhp  = (float*)ws;  ws += NC * DHID * sizeof(float); // 4 MB
  int*   u_k  = (int*)ws;    ws += NC * KNB * sizeof(int);    // 2 MB
  float* gmax = (float*)ws;  ws += NC * DHID * sizeof(float); // 4 MB  (total 36 MB)

  // 1) FPS -> p_ce (written directly into d_out tail)
  fps_kernel<<<BATCH, 1024, 0, stream>>>(p, p_ce);
  // 2) ball query -> u_ne   (one wave per center, 4 waves/block)
  ball_kernel<<<(int)(NC / 4), 128, 0, stream>>>(p, p_ce, u_ne);
  // 3) gather + maxpool -> f_ce
  pool_kernel<<<(int)NC, 128, 0, stream>>>(f, u_ne, f_ce);
  // 4) res = f_ce @ W_sc + b_sc         (tiles = 1024*16, 4 waves/block)
  gemm_wmma_kernel<<<4096, 128, 0, stream>>>(f_ce, W_sc, b_sc, nullptr, res,
                                             DOUT, DIN, 0);
  // 5) fh = lrelu(f_ce @ W_pre + b_pre) (tiles = 1024*4)
  gemm_wmma_kernel<<<1024, 128, 0, stream>>>(f_ce, W_pre, b_pre, nullptr, fhp,
                                             DHID, DIN, 1);
  // 6) kNN among centers -> u_k
  knn_kernel<<<(int)(NC / 4), 128, 0, stream>>>(p_ce, u_k);
  // 7) fused per-center MLP + max over k -> gmax
  center_kernel<<<(int)NC, 128, 0, stream>>>(p_ce, fhp, u_k, W_p2f, b_p2f,
                                             W_mlp, b_mlp, gmax);
  // 8) out = lrelu(gmax @ W_pst + b_pst + res)  (tiles = 1024*16)
  gemm_wmma_kernel<<<4096, 128, 0, stream>>>(gmax, W_pst, b_pst, res, out,
                                             DOUT, DHID, 1);
}